// SimplifiedCPELoss_53352083751398
// MI455X (gfx1250) — compile-verified
//
#include <hip/hip_runtime.h>
#include <hip/hip_bf16.h>

typedef __attribute__((ext_vector_type(2))) float v2f;
typedef __attribute__((ext_vector_type(8))) float v8f;

#define N_ROWS 8192
#define DIM 128
#define TEMP_INV 10.0f
#define NEG_INF_F (-1e30f)
#define CBLK 64          // columns staged in LDS per iteration
#define LDS_STRIDE 130   // pad 128->130 floats: keeps 8B align, avoids bank conflicts

// ---------------- Kernel 0: row-wise L2 normalize ----------------
__global__ __launch_bounds__(256) void k_normalize(const float* __restrict__ fin,
                                                   float* __restrict__ fhat) {
    const int wave = threadIdx.x >> 5;
    const int lane = threadIdx.x & 31;
    const int row  = blockIdx.x * 8 + wave;
    float4 v = ((const float4*)(fin + row * DIM))[lane];
    float ss = v.x * v.x + v.y * v.y + v.z * v.z + v.w * v.w;
#pragma unroll
    for (int m = 16; m >= 1; m >>= 1) ss += __shfl_xor(ss, m, 32);
    const float inv = 1.0f / fmaxf(sqrtf(ss), 1e-12f);
    v.x *= inv; v.y *= inv; v.z *= inv; v.w *= inv;
    ((float4*)(fhat + row * DIM))[lane] = v;
}

// ---------------- Kernels 1&2: fused sim GEMM + row statistics ----------------
// PASS==1: row-wise masked max.  PASS==2: masked exp-sums using clamped max.
template <int PASS>
__global__ __launch_bounds__(256) void k_simpass(const float* __restrict__ fhat,
                                                 const int*   __restrict__ labels,
                                                 float* __restrict__ rowMax,
                                                 float* __restrict__ rowPos,
                                                 float* __restrict__ rowAll) {
    __shared__ float ldsB[CBLK * LDS_STRIDE];
    __shared__ int   ldsLab[CBLK];

    const int tid     = threadIdx.x;
    const int wave    = tid >> 5;
    const int lane    = tid & 31;
    const int lane15  = lane & 15;
    const int half    = lane >> 4;                  // 0: lanes 0-15, 1: lanes 16-31
    const int rowBase = blockIdx.x * 128 + wave * 16;

    // A operand (16x4 f32 per WMMA): lane holds M = lane&15; K pair {k0,k0+1} for
    // lanes 0-15, {k0+2,k0+3} for lanes 16-31 (ISA 32-bit A layout). Preload all K.
    const int aRow = rowBase + lane15;
    const float* aptr = fhat + aRow * DIM + 2 * half;
    v2f areg[32];
#pragma unroll
    for (int kk = 0; kk < 32; ++kk) areg[kk] = *(const v2f*)(aptr + kk * 4);

    // C/D layout: v8f element r <-> row M = r + 8*half, col N = lane&15.
    int   rl[8];
    float mstate[8], pS[8], aS[8];
#pragma unroll
    for (int r = 0; r < 8; ++r) {
        const int i = rowBase + r + 8 * half;
        rl[r] = labels[i];
        if (PASS == 1) {
            mstate[r] = NEG_INF_F;
        } else {
            mstate[r] = fminf(fmaxf(rowMax[i], -20.0f), 20.0f);
            pS[r] = 0.0f; aS[r] = 0.0f;
        }
    }

    for (int c0 = 0; c0 < N_ROWS; c0 += CBLK) {
        __syncthreads();
        {   // cooperative stage of B block: CBLK cols x 128 features
            const int col = tid >> 2;              // 4 threads per column
            const int kb  = (tid & 3) * 32;        // 32 features each (8 x float4)
            const float* g = fhat + (c0 + col) * DIM + kb;
            float* l = ldsB + col * LDS_STRIDE + kb;
#pragma unroll
            for (int q = 0; q < 8; ++q) {
                float4 t = *(const float4*)(g + 4 * q);
                l[4*q+0] = t.x; l[4*q+1] = t.y; l[4*q+2] = t.z; l[4*q+3] = t.w;
            }
            if (tid < CBLK) ldsLab[tid] = labels[c0 + tid];
        }
        __syncthreads();

#pragma unroll
        for (int sub = 0; sub < CBLK / 16; ++sub) {
            v8f acc = {};
            // B operand (4x16 f32): lane holds N = lane&15; K pair split by half.
            const float* bptr = ldsB + (sub * 16 + lane15) * LDS_STRIDE + 2 * half;
#pragma unroll
            for (int kk = 0; kk < 32; ++kk) {
                v2f b = *(const v2f*)(bptr + kk * 4);
                acc = __builtin_amdgcn_wmma_f32_16x16x4_f32(
                    false, areg[kk], false, b, (short)0, acc, false, false);
            }

            const int j    = c0 + sub * 16 + lane15;
            const int labj = ldsLab[sub * 16 + lane15];
            const bool fgj = labj >= 0;
#pragma unroll
            for (int r = 0; r < 8; ++r) {
                const int i = rowBase + r + 8 * half;
                const float sim = acc[r] * TEMP_INV;
                const bool allm = fgj && (rl[r] >= 0) && (i != j);
                if (PASS == 1) {
                    mstate[r] = allm ? fmaxf(mstate[r], sim) : mstate[r];
                } else {
                    const float e = __expf(sim - mstate[r]);
                    aS[r] += allm ? e : 0.0f;
                    pS[r] += (allm && (rl[r] == labj)) ? e : 0.0f;
                }
            }
        }
    }

    // reduce across the 16 lanes of each half-wave (masks < 16 stay in-half)
#pragma unroll
    for (int r = 0; r < 8; ++r) {
#pragma unroll
        for (int m = 8; m >= 1; m >>= 1) {
            if (PASS == 1) {
                mstate[r] = fmaxf(mstate[r], __shfl_xor(mstate[r], m, 32));
            } else {
                pS[r] += __shfl_xor(pS[r], m, 32);
                aS[r] += __shfl_xor(aS[r], m, 32);
            }
        }
    }
    if (lane15 == 0) {
#pragma unroll
        for (int r = 0; r < 8; ++r) {
            const int i = rowBase + r + 8 * half;
            if (PASS == 1) rowMax[i] = mstate[r];
            else { rowPos[i] = pS[r]; rowAll[i] = aS[r]; }
        }
    }
}

// ---------------- Kernel 3: per-row loss + mean ----------------
__global__ __launch_bounds__(256) void k_finalize(const int* __restrict__ labels,
                                                  const float* __restrict__ rowPos,
                                                  const float* __restrict__ rowAll,
                                                  float* __restrict__ out) {
    __shared__ float sLoss[256];
    __shared__ int   sValid[256];
    __shared__ int   sFg[256];
    const int t = threadIdx.x;
    float lsum = 0.0f; int vcnt = 0, fgcnt = 0;
    for (int i = t; i < N_ROWS; i += 256) {
        const int  lab = labels[i];
        const bool fg  = lab >= 0;
        const float pr = rowPos[i], ar = rowAll[i];
        const float p  = fminf(fmaxf(pr, 1e-6f), 1e6f);
        const float a  = fminf(fmaxf(ar, 1e-6f), 1e6f);
        const float loss = fminf(-__logf(p / a), 10.0f);
        const bool valid = fg && (pr > 0.0f);   // pos_mask count > 0 <=> raw sum > 0
        if (valid) { lsum += loss; ++vcnt; }
        if (fg) ++fgcnt;
    }
    sLoss[t] = lsum; sValid[t] = vcnt; sFg[t] = fgcnt;
    __syncthreads();
    for (int s = 128; s > 0; s >>= 1) {
        if (t < s) { sLoss[t] += sLoss[t+s]; sValid[t] += sValid[t+s]; sFg[t] += sFg[t+s]; }
        __syncthreads();
    }
    if (t == 0) {
        float res = 0.0f;
        if (sFg[0] >= 2 && sValid[0] > 0) res = sLoss[0] / (float)sValid[0];
        out[0] = res;
    }
}

extern "C" void kernel_launch(void* const* d_in, const int* in_sizes, int n_in,
                              void* d_out, int out_size, void* d_ws, size_t ws_size,
                              hipStream_t stream) {
    const float* features = (const float*)d_in[0];
    const int*   labels   = (const int*)d_in[1];
    float* fhat   = (float*)d_ws;                 // 8192*128 f32 = 4 MB
    float* rowMax = fhat + N_ROWS * DIM;          // +32 KB
    float* rowPos = rowMax + N_ROWS;              // +32 KB
    float* rowAll = rowPos + N_ROWS;              // +32 KB

    k_normalize<<<N_ROWS / 8, 256, 0, stream>>>(features, fhat);
    k_simpass<1><<<N_ROWS / 128, 256, 0, stream>>>(fhat, labels, rowMax, rowPos, rowAll);
    k_simpass<2><<<N_ROWS / 128, 256, 0, stream>>>(fhat, labels, rowMax, rowPos, rowAll);
    k_finalize<<<1, 256, 0, stream>>>(labels, rowPos, rowAll, (float*)d_out);
}